// EventSceneGraph_49134425866792
// MI455X (gfx1250) — compile-verified
//
#include <hip/hip_runtime.h>
#include <hip/hip_bf16.h>
#include <math.h>

#define B   512
#define NA  128
#define NL  512
#define DD  128
#define KA  16
#define KL  16
#define LDS_LD 132          // padded row stride: banks = (4*m + k) % 64 -> conflict-free
#define NNODES (KA + KL)    // 32

typedef float v2f __attribute__((ext_vector_type(2)));
typedef float v8f __attribute__((ext_vector_type(8)));

#define NEG_INF (-__builtin_huge_valf())
#define POS_INF ( __builtin_huge_valf())

// ---------------------------------------------------------------------------
// Kernel 1: masked top-KA agents per batch (ties -> lower index, matching
// jax.lax.top_k). One block (128 threads) per batch.
// ---------------------------------------------------------------------------
__global__ __launch_bounds__(NA)
void esg_topk_agents(const float* __restrict__ spike,
                     const unsigned char* __restrict__ valid,
                     int* __restrict__ aidx_out) {
    const int b = blockIdx.x;
    const int t = threadIdx.x;
    __shared__ float vals[NA];
    __shared__ float rv[NA];
    __shared__ int   ri[NA];

    vals[t] = valid[b * NA + t] ? spike[b * NA + t] : NEG_INF;
    __syncthreads();

    for (int it = 0; it < KA; ++it) {
        rv[t] = vals[t];
        ri[t] = t;
        __syncthreads();
        for (int s = NA / 2; s > 0; s >>= 1) {
            if (t < s) {
                float va = rv[t], vb = rv[t + s];
                int   ia = ri[t], ib = ri[t + s];
                if (vb > va || (vb == va && ib < ia)) { rv[t] = vb; ri[t] = ib; }
            }
            __syncthreads();
        }
        if (t == 0) {
            aidx_out[b * KA + it] = ri[0];
            vals[ri[0]] = NEG_INF;     // remove winner
        }
        __syncthreads();
    }
}

// ---------------------------------------------------------------------------
// Kernel 2: lane selection. dist[l] = min_k ||ac_k - lane_l||, invalid -> +inf,
// pick KL smallest (ties -> lower index). One block (512 threads) per batch.
// ---------------------------------------------------------------------------
__global__ __launch_bounds__(NL)
void esg_lane_select(const float* __restrict__ lane_centers,
                     const float* __restrict__ x_centers,
                     const unsigned char* __restrict__ lane_valid,
                     const int* __restrict__ aidx,
                     int* __restrict__ lidx_out) {
    const int b = blockIdx.x;
    const int t = threadIdx.x;
    __shared__ float acx[KA], acy[KA];
    __shared__ float dist[NL];
    __shared__ float rv[NL];
    __shared__ int   ri[NL];

    if (t < KA) {
        int a = aidx[b * KA + t];
        acx[t] = x_centers[((size_t)b * NA + a) * 2 + 0];
        acy[t] = x_centers[((size_t)b * NA + a) * 2 + 1];
    }
    __syncthreads();

    float lx = lane_centers[((size_t)b * NL + t) * 2 + 0];
    float ly = lane_centers[((size_t)b * NL + t) * 2 + 1];
    float dmin = POS_INF;
#pragma unroll
    for (int k = 0; k < KA; ++k) {
        float dx = acx[k] - lx, dy = acy[k] - ly;
        dmin = fminf(dmin, sqrtf(dx * dx + dy * dy));
    }
    if (!lane_valid[b * NL + t]) dmin = POS_INF;
    dist[t] = dmin;
    __syncthreads();

    for (int it = 0; it < KL; ++it) {
        rv[t] = dist[t];
        ri[t] = t;
        __syncthreads();
        for (int s = NL / 2; s > 0; s >>= 1) {
            if (t < s) {
                float va = rv[t], vb = rv[t + s];
                int   ia = ri[t], ib = ri[t + s];
                if (vb < va || (vb == va && ib < ia)) { rv[t] = vb; ri[t] = ib; }
            }
            __syncthreads();
        }
        if (t == 0) {
            lidx_out[b * KL + it] = ri[0];
            dist[ri[0]] = POS_INF;
        }
        __syncthreads();
    }
}

// ---------------------------------------------------------------------------
// Kernel 3: gather -> 2x (residual MLP + LayerNorm) on WMMA f32 -> scatter.
// One block (512 thr = 16 waves) per batch. Wave w computes output tile
// (mt = w/8 in {0,1}) x (nt = w%8 in 0..7) of the [32 x 128] activations.
// A (nodes/hidden) from LDS, B (weights) streamed from L2.
// ---------------------------------------------------------------------------
__device__ __forceinline__ v8f esg_gemm_tile(const float* __restrict__ sA,
                                             const float* __restrict__ W,
                                             int mrow, int koff, int ncol) {
    v8f acc = {};
#pragma unroll 8
    for (int k = 0; k < DD; k += 4) {
        v2f a, bf;
        a.x  = sA[mrow * LDS_LD + k + koff];
        a.y  = sA[mrow * LDS_LD + k + koff + 1];
        bf.x = W[(k + koff) * DD + ncol];
        bf.y = W[(k + koff + 1) * DD + ncol];
        acc = __builtin_amdgcn_wmma_f32_16x16x4_f32(
            /*neg_a=*/false, a, /*neg_b=*/false, bf,
            /*c_mod=*/(short)0, acc, /*reuse_a=*/false, /*reuse_b=*/false);
    }
    return acc;
}

__global__ __launch_bounds__(512)
void esg_mlp(const float* __restrict__ actor_feat,
             const float* __restrict__ lane_feat,
             const float* __restrict__ W0a, const float* __restrict__ b0a,
             const float* __restrict__ W0b, const float* __restrict__ b0b,
             const float* __restrict__ W1a, const float* __restrict__ b1a,
             const float* __restrict__ W1b, const float* __restrict__ b1b,
             const float* __restrict__ gamma, const float* __restrict__ beta,
             const int* __restrict__ aidx, const int* __restrict__ lidx,
             float* __restrict__ actor_out, float* __restrict__ lane_out) {
    const int b = blockIdx.x;
    const int t = threadIdx.x;

    __shared__ float sN[NNODES * LDS_LD];   // current node activations
    __shared__ float sH[NNODES * LDS_LD];   // hidden (post-gelu)
    __shared__ float sSum[NNODES * 16];
    __shared__ float sSq[NNODES * 16];
    __shared__ float sMean[NNODES];
    __shared__ float sRstd[NNODES];
    __shared__ int   sRow[NNODES];

    const int r  = t >> 4;          // node row 0..31 (copy/LN mapping)
    const int cc = (t & 15) * 8;    // 8 contiguous cols per thread

    if (t < NNODES)
        sRow[t] = (t < KA) ? aidx[b * KA + t] : lidx[b * KL + (t - KA)];
    __syncthreads();

    {   // gather 32 x 128 nodes into LDS
        const float* src = (r < KA)
            ? (actor_feat + ((size_t)b * NA + sRow[r]) * DD)
            : (lane_feat  + ((size_t)b * NL + sRow[r]) * DD);
#pragma unroll
        for (int i = 0; i < 8; ++i) sN[r * LDS_LD + cc + i] = src[cc + i];
    }
    __syncthreads();

    const int wave = t >> 5;
    const int lane = t & 31;
    const int mt   = wave >> 3;                 // 0..1
    const int nt   = wave & 7;                  // 0..7
    const int mrow = mt * 16 + (lane & 15);     // A-fragment row
    const int koff = (lane >> 4) * 2;           // 0 or 2
    const int ncol = nt * 16 + (lane & 15);     // output column
    const int rhalf = (lane >> 4) * 8;          // C/D row offset of lane half

    const float* Wa[2] = {W0a, W1a};
    const float* Ba[2] = {b0a, b1a};
    const float* Wb[2] = {W0b, W1b};
    const float* Bb[2] = {b0b, b1b};

    for (int layer = 0; layer < 2; ++layer) {
        // ---- H = gelu(nodes @ Wa + ba) ----
        v8f acc = esg_gemm_tile(sN, Wa[layer], mrow, koff, ncol);
        float bias = Ba[layer][ncol];
#pragma unroll
        for (int j = 0; j < 8; ++j) {
            int row = mt * 16 + j + rhalf;
            float x = acc[j] + bias;
            float g = 0.5f * x * (1.0f + erff(x * 0.70710678118654752f));
            sH[row * LDS_LD + nt * 16 + (lane & 15)] = g;
        }
        __syncthreads();

        // ---- h = H @ Wb + bb ; nodes += h (each element owned by 1 lane) ----
        v8f acc2 = esg_gemm_tile(sH, Wb[layer], mrow, koff, ncol);
        float bias2 = Bb[layer][ncol];
#pragma unroll
        for (int j = 0; j < 8; ++j) {
            int row = mt * 16 + j + rhalf;
            int col = nt * 16 + (lane & 15);
            sN[row * LDS_LD + col] += acc2[j] + bias2;
        }
        __syncthreads();

        // ---- LayerNorm over D per row (16 threads x 8 elems per row) ----
        float s0 = 0.f, s1 = 0.f;
#pragma unroll
        for (int i = 0; i < 8; ++i) {
            float x = sN[r * LDS_LD + cc + i];
            s0 += x; s1 += x * x;
        }
        sSum[r * 16 + (t & 15)] = s0;
        sSq [r * 16 + (t & 15)] = s1;
        __syncthreads();
        if ((t & 15) == 0) {
            float m0 = 0.f, m1 = 0.f;
            for (int i = 0; i < 16; ++i) { m0 += sSum[r * 16 + i]; m1 += sSq[r * 16 + i]; }
            float mean = m0 * (1.0f / DD);
            float var  = m1 * (1.0f / DD) - mean * mean;
            sMean[r] = mean;
            sRstd[r] = rsqrtf(var + 1e-5f);
        }
        __syncthreads();
        float mean = sMean[r], rstd = sRstd[r];
#pragma unroll
        for (int i = 0; i < 8; ++i) {
            int c = cc + i;
            sN[r * LDS_LD + c] = (sN[r * LDS_LD + c] - mean) * rstd * gamma[c] + beta[c];
        }
        __syncthreads();
    }

    {   // scatter updated nodes into full pools (indices are distinct)
        float* dst = (r < KA)
            ? (actor_out + ((size_t)b * NA + sRow[r]) * DD)
            : (lane_out  + ((size_t)b * NL + sRow[r]) * DD);
#pragma unroll
        for (int i = 0; i < 8; ++i) dst[cc + i] = sN[r * LDS_LD + cc + i];
    }
}

// ---------------------------------------------------------------------------
extern "C" void kernel_launch(void* const* d_in, const int* in_sizes, int n_in,
                              void* d_out, int out_size, void* d_ws, size_t ws_size,
                              hipStream_t stream) {
    const float* actor_feat   = (const float*)d_in[0];
    const float* lane_feat    = (const float*)d_in[1];
    const float* lane_centers = (const float*)d_in[2];
    const float* x_centers    = (const float*)d_in[3];
    const float* spike_rate   = (const float*)d_in[4];
    const unsigned char* actor_valid = (const unsigned char*)d_in[5]; // jax bool = 1 byte
    const unsigned char* lane_valid  = (const unsigned char*)d_in[6];
    const float* W0a = (const float*)d_in[7];
    const float* b0a = (const float*)d_in[8];
    const float* W0b = (const float*)d_in[9];
    const float* b0b = (const float*)d_in[10];
    const float* W1a = (const float*)d_in[11];
    const float* b1a = (const float*)d_in[12];
    const float* W1b = (const float*)d_in[13];
    const float* b1b = (const float*)d_in[14];
    const float* gamma = (const float*)d_in[15];
    const float* beta  = (const float*)d_in[16];

    float* actor_out = (float*)d_out;                       // [B,NA,D]
    float* lane_out  = actor_out + (size_t)B * NA * DD;     // [B,NL,D]

    int* aidx = (int*)d_ws;          // [B,KA]
    int* lidx = aidx + B * KA;       // [B,KL]

    // Bulk passthrough of both pools (HBM-bandwidth bound, ~336 MB round trip).
    hipMemcpyAsync(actor_out, actor_feat, (size_t)B * NA * DD * sizeof(float),
                   hipMemcpyDeviceToDevice, stream);
    hipMemcpyAsync(lane_out, lane_feat, (size_t)B * NL * DD * sizeof(float),
                   hipMemcpyDeviceToDevice, stream);

    esg_topk_agents<<<B, NA, 0, stream>>>(spike_rate, actor_valid, aidx);
    esg_lane_select<<<B, NL, 0, stream>>>(lane_centers, x_centers, lane_valid,
                                          aidx, lidx);
    esg_mlp<<<B, 512, 0, stream>>>(actor_feat, lane_feat,
                                   W0a, b0a, W0b, b0b, W1a, b1a, W1b, b1b,
                                   gamma, beta, aidx, lidx,
                                   actor_out, lane_out);
}